// SoftMaxMin_9749575762015
// MI455X (gfx1250) — compile-verified
//
#include <hip/hip_runtime.h>
#include <math.h>

typedef __attribute__((ext_vector_type(2))) float v2f;
typedef __attribute__((ext_vector_type(8))) float v8f;

#define B_ROWS 2048
#define F_DIM  512
#define U_DIM  256
#define NU_    128          // units per sign group
#define BETA_  15.0f

#define MT_   32            // output rows per block
#define KC_   64            // K-chunk staged in LDS
#define XPAD  516           // LDS row stride for X tile (512 + 4 pad -> conflict-free)

// Packed B layout: one float4 per (K-group g, half-wave hk, strip st, lane l16):
//   ( W[4g+hk][c], W[4g+2+hk][c], W[4g+hk][c+16], W[4g+2+hk][c+16] ),  c = 32*st + l16
// float4 index = (g*2 + hk)*64 + st*16 + l16.  Group stride = 2 KB (16-bit DS imm),
// each half-wave reads a contiguous 256 B span -> conflict-free ds_load_b128.
#define PW_CHUNK_F4      ((KC_ / 4) * 2 * 4 * 16)            // 2048 float4 per chunk
#define PW_CHUNK_FLOATS  (PW_CHUNK_F4 * 4)                   // 8192 floats = 32 KB
#define PW_SIGN_FLOATS   ((F_DIM / KC_) * PW_CHUNK_FLOATS)   // 8 chunks per sign

// ---------------------------------------------------------------------------
// Preprocess W: per unit u, m = max_f(sign*beta*w[f,u]); emit exp(v - m)
// directly in the packed B-fragment chunk layout.  One wave per unit.
// ---------------------------------------------------------------------------
__global__ __launch_bounds__(32) void prep_w_kernel(const float* __restrict__ w,
                                                    float* __restrict__ ewpk,
                                                    float* __restrict__ colmax) {
    const int u    = blockIdx.x;            // 0..255
    const int lane = threadIdx.x;           // 0..31
    const int sign = (u >= NU_) ? 1 : 0;
    const int n    = u & (NU_ - 1);
    const float bs = sign ? -BETA_ : BETA_;

    const int st     = n >> 5;              // 32-column strip
    const int half16 = (n >> 4) & 1;        // fragment (n vs n+16) within strip
    const int l16n   = n & 15;              // lane within fragment

    float vals[F_DIM / 32];
    float m = -INFINITY;
#pragma unroll
    for (int i = 0; i < F_DIM / 32; ++i) {
        const int f = lane + i * 32;
        const float v = bs * w[f * U_DIM + u];
        vals[i] = v;
        m = fmaxf(m, v);
    }
#pragma unroll
    for (int off = 16; off >= 1; off >>= 1)
        m = fmaxf(m, __shfl_xor(m, off, 32));

    float* dst = ewpk + (size_t)sign * PW_SIGN_FLOATS;
#pragma unroll
    for (int i = 0; i < F_DIM / 32; ++i) {
        const int f   = lane + i * 32;
        const int c   = f >> 6;              // chunk
        const int kin = f & (KC_ - 1);
        const int g   = kin >> 2;            // K-group within chunk
        const int hk  = kin & 1;             // half-wave slot
        const int sl  = (kin >> 1) & 1;      // K / K+2 within the pair
        const size_t fi =
            (size_t)c * PW_CHUNK_FLOATS +
            (size_t)(((g * 2 + hk) * 64 + st * 16 + l16n) * 4 + half16 * 2 + sl);
        dst[fi] = __expf(vals[i] - m);
    }
    if (lane == 0) colmax[u] = m;
}

// ---------------------------------------------------------------------------
// Fused kernel: out[b, sign*128 + n] = s * (mx[b] + mw[u] + log(Ex @ Ew)) / beta
// Block: 256 threads = 8 wave32; tile = 32 (M) x 128 (N); K = 512 in LDS chunks.
// Inner loop per K-step: 0.5x ds_load_2addr_b64 (A) + 1x ds_load_b128 (B)
// + 2x v_wmma_f32_16x16x4_f32, immediate-offset addressing.
// ---------------------------------------------------------------------------
__global__ __launch_bounds__(256) void lse_gemm_kernel(const float* __restrict__ x,
                                                       const float* __restrict__ ewpk,
                                                       const float* __restrict__ colmax,
                                                       float* __restrict__ out) {
    __shared__ float sX[MT_ * XPAD];        // exp(sign*beta*x - rowmax), padded rows
    __shared__ float sW[PW_CHUNK_FLOATS];   // packed B-fragment chunk image
    __shared__ float sMax[MT_];             // per-row max of sign*beta*x

    const int tid   = threadIdx.x;
    const int wave  = tid >> 5;             // 0..7
    const int lane  = tid & 31;
    const int l16   = lane & 15;
    const int halfk = lane >> 4;            // 0 or 1 (half-wave)

    const int sign    = blockIdx.y;         // 0: smooth-max half, 1: smooth-min half
    const int rowBase = blockIdx.x * MT_;
    const float bs    = sign ? -BETA_ : BETA_;
    const float* ewp  = ewpk + (size_t)sign * PW_SIGN_FLOATS;

    // ---- Stage A: row-max + exp of the x tile into LDS (wave w -> rows 4w..4w+3)
#pragma unroll
    for (int rr = 0; rr < 4; ++rr) {
        const int row = wave * 4 + rr;
        const float* xr = x + (size_t)(rowBase + row) * F_DIM;
        float vals[F_DIM / 32];
        float m = -INFINITY;
#pragma unroll
        for (int i = 0; i < F_DIM / 32; ++i) {
            const float v = bs * xr[lane + i * 32];
            vals[i] = v;
            m = fmaxf(m, v);
        }
#pragma unroll
        for (int off = 16; off >= 1; off >>= 1)
            m = fmaxf(m, __shfl_xor(m, off, 32));
        if (lane == 0) sMax[row] = m;
#pragma unroll
        for (int i = 0; i < F_DIM / 32; ++i)
            sX[row * XPAD + lane + i * 32] = __expf(vals[i] - m);
    }
    __syncthreads();

    // ---- Stage B: K-chunked WMMA GEMM
    const int mOff = (wave & 1) * 16;       // M offset of this wave's strip
    const int nOff = (wave >> 1) * 32;      // N offset (strip) of this wave

    v8f acc0 = {0.f, 0.f, 0.f, 0.f, 0.f, 0.f, 0.f, 0.f};
    v8f acc1 = {0.f, 0.f, 0.f, 0.f, 0.f, 0.f, 0.f, 0.f};

    for (int kc = 0; kc < F_DIM; kc += KC_) {
        // straight float4 memcpy of the pre-packed chunk image into LDS
        const float* wsrc = ewp + (size_t)(kc / KC_) * PW_CHUNK_FLOATS;
#pragma unroll
        for (int i = 0; i < PW_CHUNK_FLOATS / (256 * 4); ++i) {   // 8 x b128
            const int e = (tid + i * 256) * 4;
            *(float4*)&sW[e] = *(const float4*)(wsrc + e);
        }
        // speculative prefetch of next chunk into GL2 (global_prefetch_b8)
        if (kc + KC_ < F_DIM)
            __builtin_prefetch(wsrc + PW_CHUNK_FLOATS + tid * 32, 0, 0);
        __syncthreads();

        const float*  xrow  = &sX[(mOff + l16) * XPAD + kc];
        const float4* bbase = (const float4*)sW + (halfk * 64 + (nOff >> 5) * 16 + l16);
#pragma unroll 8
        for (int k0 = 0; k0 < KC_; k0 += 4) {
            // A fragment (16x4 f32): lane M=l16; half-wave holds K {k0,k0+1}/{k0+2,k0+3}
            v2f a;
            a.x = xrow[k0 + 2 * halfk];
            a.y = xrow[k0 + 2 * halfk + 1];
            // B fragments: one b128 per K-step, immediate offset (2 KB group stride)
            const float4 bb = bbase[(k0 >> 2) * 128];
            v2f b0; b0.x = bb.x; b0.y = bb.y;
            v2f b1; b1.x = bb.z; b1.y = bb.w;

            acc0 = __builtin_amdgcn_wmma_f32_16x16x4_f32(false, a, false, b0,
                                                         (short)0, acc0, false, false);
            acc1 = __builtin_amdgcn_wmma_f32_16x16x4_f32(false, a, false, b1,
                                                         (short)0, acc1, false, false);
        }
        __syncthreads();
    }

    // ---- Stage C: epilogue  out = s * (rowmax + colmax + log(acc)) / beta
    const float inv = 1.0f / BETA_;
    const float s   = sign ? -1.0f : 1.0f;
#pragma unroll
    for (int i = 0; i < 8; ++i) {
        const int r  = mOff + i + 8 * halfk;          // C/D layout: VGPR i, half-wave
        const int c0 = nOff + l16;
        const int c1 = c0 + 16;
        const int u0 = sign * NU_ + c0;
        const int u1 = sign * NU_ + c1;
        const float rm = sMax[r];
        float* orow = out + (size_t)(rowBase + r) * U_DIM;
        orow[u0] = s * (rm + colmax[u0] + __logf(acc0[i])) * inv;
        orow[u1] = s * (rm + colmax[u1] + __logf(acc1[i])) * inv;
    }
}

// ---------------------------------------------------------------------------
extern "C" void kernel_launch(void* const* d_in, const int* in_sizes, int n_in,
                              void* d_out, int out_size, void* d_ws, size_t ws_size,
                              hipStream_t stream) {
    (void)in_sizes; (void)n_in; (void)out_size; (void)ws_size;
    const float* x = (const float*)d_in[0];     // (2048, 512)
    const float* w = (const float*)d_in[1];     // (512, 256)
    float* out = (float*)d_out;                 // (2048, 256)

    float* ws     = (float*)d_ws;
    float* ewpk   = ws;                              // 2 * PW_SIGN_FLOATS floats
    float* colmax = ws + 2 * (size_t)PW_SIGN_FLOATS; // 256 floats

    prep_w_kernel<<<U_DIM, 32, 0, stream>>>(w, ewpk, colmax);

    dim3 grid(B_ROWS / MT_, 2);
    lse_gemm_kernel<<<grid, 256, 0, stream>>>(x, ewpk, colmax, out);
}